// ToxicityGNN_7988639171338
// MI455X (gfx1250) — compile-verified
//
#include <hip/hip_runtime.h>
#include <hip/hip_bf16.h>

#define NN 100000
#define NE 1600000
#define HID 128
#define NG 2048
#define MDIM 64

typedef float v2f __attribute__((ext_vector_type(2)));
typedef float v8f __attribute__((ext_vector_type(8)));

// ---------------- utility ----------------
__global__ void fill_kernel(float* __restrict__ p, float v, int n) {
    int i = blockIdx.x * blockDim.x + threadIdx.x;
    if (i < n) p[i] = v;
}

__global__ void deg_accum_kernel(const int* __restrict__ dst, float* __restrict__ deg) {
    int e = blockIdx.x * blockDim.x + threadIdx.x;
    if (e < NE) atomicAdd(&deg[dst[e]], 1.0f);
}

// deg (in place) -> dinv ; self_norm = 1/deg
__global__ void norm_kernel(float* __restrict__ deg_io, float* __restrict__ self_norm, int n) {
    int i = blockIdx.x * blockDim.x + threadIdx.x;
    if (i < n) {
        float d = deg_io[i];
        deg_io[i] = rsqrtf(d);
        self_norm[i] = 1.0f / d;
    }
}

// ---------------- GEMM: H = X @ W ; AGG = H*self_norm + b (fused epilogue) ------------
// grid.x = NN/16 = 6250 blocks, 256 threads (8 waves). Wave w computes the
// 16x16 tile at rows [blockIdx*16,+16), cols [w*16,+16). K=128 via 32x
// v_wmma_f32_16x16x4_f32 (exact fp32).
__global__ __launch_bounds__(256) void gcn_gemm_kernel(
    const float* __restrict__ X, const float* __restrict__ W,
    const float* __restrict__ bias, const float* __restrict__ self_norm,
    float* __restrict__ H, float* __restrict__ AGG) {
    const int lane = threadIdx.x & 31;
    const int wave = threadIdx.x >> 5;
    const int m0 = blockIdx.x * 16;
    const int n0 = wave * 16;
    const int half = lane >> 4;          // 0: lanes 0-15, 1: lanes 16-31
    const int kOff = half * 2;           // K sub-offset for A and B fragments
    const int mA = lane & 15;            // A: lane carries row M
    const int nB = lane & 15;            // B/C/D: lane carries col N

    // A 16x4 f32: VGPR0 = K=kOff, VGPR1 = K=kOff+1 (per half-wave)
    const float* arow = X + (size_t)(m0 + mA) * HID + kOff;
    // B 4x16 f32: VGPR0 = row K=kOff, VGPR1 = row K=kOff+1
    const float* bcol = W + (size_t)kOff * HID + n0 + nB;

    v8f c = {0.f, 0.f, 0.f, 0.f, 0.f, 0.f, 0.f, 0.f};
#pragma unroll
    for (int k = 0; k < HID; k += 4) {
        v2f a = *(const v2f*)(arow + k);                 // 8B-aligned contiguous
        v2f b;
        b.x = bcol[(size_t)k * HID];
        b.y = bcol[(size_t)(k + 1) * HID];
        c = __builtin_amdgcn_wmma_f32_16x16x4_f32(
            /*neg_a=*/false, a, /*neg_b=*/false, b,
            /*c_mod=*/(short)0, c, /*reuse_a=*/false, /*reuse_b=*/false);
    }

    // C/D 16x16 f32: VGPR i -> M = half*8 + i, N = lane&15
    const int n = n0 + nB;
    const int mBase = m0 + half * 8;
    const float bn = bias[n];
#pragma unroll
    for (int i = 0; i < 8; ++i) {
        int m = mBase + i;
        float v = c[i];
        H[(size_t)m * HID + n] = v;
        AGG[(size_t)m * HID + n] = fmaf(v, self_norm[m], bn);
    }
}

// ---------------- edge scatter: AGG[dst] += H[src] * dinv[src]*dinv[dst] -----------
// one wave per edge; lane handles 4 contiguous floats (32*4 = 128 = full row)
__global__ __launch_bounds__(256) void gcn_scatter_kernel(
    const float* __restrict__ H, const int* __restrict__ src,
    const int* __restrict__ dst, const float* __restrict__ dinv,
    float* __restrict__ AGG) {
    int e = blockIdx.x * 8 + (threadIdx.x >> 5);
    if (e >= NE) return;
    int lane = threadIdx.x & 31;
    int s = src[e], d = dst[e];
    float en = dinv[s] * dinv[d];
    const float4 v = *(const float4*)(H + (size_t)s * HID + lane * 4);
    float* p = AGG + (size_t)d * HID + lane * 4;
    atomicAdd(p + 0, v.x * en);
    atomicAdd(p + 1, v.y * en);
    atomicAdd(p + 2, v.z * en);
    atomicAdd(p + 3, v.w * en);
}

__global__ void relu_copy_kernel(const float* __restrict__ in, float* __restrict__ out, int n) {
    int i = blockIdx.x * blockDim.x + threadIdx.x;
    if (i < n) out[i] = fmaxf(in[i], 0.0f);
}

// ---------------- pooling: wave per node, float4 per lane ----------------
__global__ __launch_bounds__(256) void pool_accum_kernel(
    const float* __restrict__ X, const int* __restrict__ batch,
    float* __restrict__ sums, unsigned int* __restrict__ maxb,
    float* __restrict__ counts) {
    int node = blockIdx.x * 8 + (threadIdx.x >> 5);
    if (node >= NN) return;
    int lane = threadIdx.x & 31;
    int g = batch[node];
    const float4 v = *(const float4*)(X + (size_t)node * HID + lane * 4);
    float* sp = sums + (size_t)g * HID + lane * 4;
    unsigned int* mp = maxb + (size_t)g * HID + lane * 4;
    atomicAdd(sp + 0, v.x);
    atomicAdd(sp + 1, v.y);
    atomicAdd(sp + 2, v.z);
    atomicAdd(sp + 3, v.w);
    // values are post-ReLU (>= 0): uint bit order == float order
    atomicMax(mp + 0, __float_as_uint(v.x));
    atomicMax(mp + 1, __float_as_uint(v.y));
    atomicMax(mp + 2, __float_as_uint(v.z));
    atomicMax(mp + 3, __float_as_uint(v.w));
    if (lane == 0) atomicAdd(&counts[g], 1.0f);
}

// fused[g, 0:128]=mean, [128:256]=max, [256:288]=relu(meta@Wm+bm)
__global__ __launch_bounds__(288) void fuse_kernel(
    const float* __restrict__ sums, const float* __restrict__ maxb,
    const float* __restrict__ counts, const float* __restrict__ metadata,
    const float* __restrict__ Wm, const float* __restrict__ bm,
    float* __restrict__ fused) {
    int g = blockIdx.x;
    int t = threadIdx.x;
    float* frow = fused + (size_t)g * 288;
    if (t < HID) {
        float cnt = fmaxf(counts[g], 1.0f);
        frow[t] = sums[(size_t)g * HID + t] / cnt;
    } else if (t < 2 * HID) {
        frow[t] = maxb[(size_t)g * HID + (t - HID)];
    } else {
        int j = t - 2 * HID;  // 0..31
        float acc = bm[j];
        const float* mrow = metadata + (size_t)g * MDIM;
        for (int k = 0; k < MDIM; ++k) acc = fmaf(mrow[k], Wm[k * 32 + j], acc);
        frow[t] = fmaxf(acc, 0.0f);
    }
}

// head: hidden = relu(fused @ Wp1 + bp1); out = hidden @ Wp2 + bp2
__global__ __launch_bounds__(64) void head_kernel(
    const float* __restrict__ fused, const float* __restrict__ Wp1,
    const float* __restrict__ bp1, const float* __restrict__ Wp2,
    const float* __restrict__ bp2, float* __restrict__ out) {
    __shared__ float hid[64];
    int g = blockIdx.x;
    int j = threadIdx.x;
    const float* frow = fused + (size_t)g * 288;
    float acc = bp1[j];
    for (int k = 0; k < 288; ++k) acc = fmaf(frow[k], Wp1[k * 64 + j], acc);
    hid[j] = fmaxf(acc, 0.0f);
    __syncthreads();
    if (j == 0) {
        float o = bp2[0];
        for (int k = 0; k < 64; ++k) o = fmaf(hid[k], Wp2[k], o);
        out[g] = o;
    }
}

// ---------------- launch ----------------
extern "C" void kernel_launch(void* const* d_in, const int* in_sizes, int n_in,
                              void* d_out, int out_size, void* d_ws, size_t ws_size,
                              hipStream_t stream) {
    const float* x        = (const float*)d_in[0];
    const int*   ei       = (const int*)d_in[1];   // [2, NE]
    const int*   batch    = (const int*)d_in[2];
    const float* metadata = (const float*)d_in[3];
    const float* W1 = (const float*)d_in[4];
    const float* b1 = (const float*)d_in[5];
    const float* W2 = (const float*)d_in[6];
    const float* b2 = (const float*)d_in[7];
    const float* W3 = (const float*)d_in[8];
    const float* b3 = (const float*)d_in[9];
    const float* Wm = (const float*)d_in[10];
    const float* bm = (const float*)d_in[11];
    const float* Wp1 = (const float*)d_in[12];
    const float* bp1 = (const float*)d_in[13];
    const float* Wp2 = (const float*)d_in[14];
    const float* bp2 = (const float*)d_in[15];
    float* out = (float*)d_out;

    const int* src = ei;
    const int* dst = ei + NE;

    // workspace layout (floats)
    float* ws = (float*)d_ws;
    const size_t NF = (size_t)NN * HID;        // 12,800,000
    float* H     = ws;                         // [NN,128]
    float* AGG   = H + NF;                     // [NN,128]
    float* X1    = AGG + NF;                   // [NN,128]
    float* dinv  = X1 + NF;                    // [NN] (first holds deg)
    float* selfn = dinv + NN;                  // [NN]
    float* sums  = selfn + NN;                 // [NG,128]
    float* maxb  = sums + (size_t)NG * HID;    // [NG,128]
    float* cnts  = maxb + (size_t)NG * HID;    // [NG]
    float* fused = cnts + NG;                  // [NG,288]

    const int T = 256;
    // degrees / norms
    fill_kernel<<<(NN + T - 1) / T, T, 0, stream>>>(dinv, 1.0f, NN);
    deg_accum_kernel<<<(NE + T - 1) / T, T, 0, stream>>>(dst, dinv);
    norm_kernel<<<(NN + T - 1) / T, T, 0, stream>>>(dinv, selfn, NN);

    const int gemm_blocks = NN / 16;           // 6250, exact
    const int scat_blocks = NE / 8;            // 200000, exact
    const int relu_blocks = (int)(NF / T);     // 50000, exact

    // layer 1
    gcn_gemm_kernel<<<gemm_blocks, 256, 0, stream>>>(x, W1, b1, selfn, H, AGG);
    gcn_scatter_kernel<<<scat_blocks, 256, 0, stream>>>(H, src, dst, dinv, AGG);
    relu_copy_kernel<<<relu_blocks, T, 0, stream>>>(AGG, X1, (int)NF);
    // layer 2
    gcn_gemm_kernel<<<gemm_blocks, 256, 0, stream>>>(X1, W2, b2, selfn, H, AGG);
    gcn_scatter_kernel<<<scat_blocks, 256, 0, stream>>>(H, src, dst, dinv, AGG);
    relu_copy_kernel<<<relu_blocks, T, 0, stream>>>(AGG, X1, (int)NF);
    // layer 3
    gcn_gemm_kernel<<<gemm_blocks, 256, 0, stream>>>(X1, W3, b3, selfn, H, AGG);
    gcn_scatter_kernel<<<scat_blocks, 256, 0, stream>>>(H, src, dst, dinv, AGG);
    relu_copy_kernel<<<relu_blocks, T, 0, stream>>>(AGG, X1, (int)NF);

    // pooling
    const int pf = NG * HID;
    fill_kernel<<<(pf + T - 1) / T, T, 0, stream>>>(sums, 0.0f, pf);
    fill_kernel<<<(pf + T - 1) / T, T, 0, stream>>>(maxb, 0.0f, pf);
    fill_kernel<<<(NG + T - 1) / T, T, 0, stream>>>(cnts, 0.0f, NG);
    pool_accum_kernel<<<NN / 8, 256, 0, stream>>>(X1, batch, sums, (unsigned int*)maxb, cnts);

    fuse_kernel<<<NG, 288, 0, stream>>>(sums, maxb, cnts, metadata, Wm, bm, fused);
    head_kernel<<<NG, 64, 0, stream>>>(fused, Wp1, bp1, Wp2, bp2, out);
}